// Ive_42417097016195
// MI455X (gfx1250) — compile-verified
//
#include <hip/hip_runtime.h>

// i1e(x) = I1(x) * exp(-|x|), Abramowitz & Stegun 9.8.3/9.8.4 polynomials,
// matching the JAX reference in structure (fp32, FMA Horner).
//
// MI455X reasoning: 33.55M fp32 elements -> 256 MiB total traffic, HBM-bound
// (~11.5 us floor @ 23.3 TB/s). Strategy: b128 vector loads/stores with
// non-temporal hints (working set > 192MB L2, zero reuse), hardware
// transcendentals (v_exp_f32 / v_rcp_f32 / v_rsq_f32), branchless select.

// Native clang vector type: accepted by __builtin_nontemporal_* and lowers
// to a single 128-bit global memory op.
typedef float v4f __attribute__((ext_vector_type(4)));

__device__ __forceinline__ float i1e_eval(float x) {
    const float ax = __builtin_fabsf(x);

    // ---- small branch: ax < 3.75, t = (ax/3.75)^2 ----
    float t = ax * (1.0f / 3.75f);
    t = t * t;
    float ps = 0.00032411f;
    ps = __builtin_fmaf(ps, t, 0.00301532f);
    ps = __builtin_fmaf(ps, t, 0.02658733f);
    ps = __builtin_fmaf(ps, t, 0.15084934f);
    ps = __builtin_fmaf(ps, t, 0.51498869f);
    ps = __builtin_fmaf(ps, t, 0.87890594f);
    ps = __builtin_fmaf(ps, t, 0.5f);
    // exp(-ax) = exp2(-ax * log2(e)) -> single v_exp_f32
    const float e = __builtin_amdgcn_exp2f(ax * -1.4426950408889634f);
    const float small = x * ps * e;

    // ---- large branch: ax >= 3.75, tl = 3.75 / ax ----
    const float m  = __builtin_fmaxf(ax, 3.75f);        // safe denom for unused path
    const float tl = 3.75f * __builtin_amdgcn_rcpf(m);  // v_rcp_f32
    float pl = -0.00420059f;
    pl = __builtin_fmaf(pl, tl,  0.01787654f);
    pl = __builtin_fmaf(pl, tl, -0.02895312f);
    pl = __builtin_fmaf(pl, tl,  0.02282967f);
    pl = __builtin_fmaf(pl, tl, -0.01031555f);
    pl = __builtin_fmaf(pl, tl,  0.00163801f);
    pl = __builtin_fmaf(pl, tl, -0.00362018f);
    pl = __builtin_fmaf(pl, tl, -0.03988024f);
    pl = __builtin_fmaf(pl, tl,  0.39894228f);
    const float rs  = __builtin_amdgcn_rsqf(m);         // v_rsq_f32: 1/sqrt(m)
    const float sgn = __builtin_copysignf(1.0f, x);     // x==0 resolves via small branch
    const float large = pl * rs * sgn;

    return (ax < 3.75f) ? small : large;                // v_cndmask, branchless
}

__global__ void __launch_bounds__(256)
ive_i1e_kernel(const float* __restrict__ z, float* __restrict__ out, long long n) {
    const long long n4   = n >> 2;
    const long long gtid = (long long)blockIdx.x * blockDim.x + threadIdx.x;

    if (gtid < n4) {
        // 128-bit NT load -> global_load_b128 TH=NT (stream, no L2 residency)
        const v4f* zin = reinterpret_cast<const v4f*>(z);
        v4f*       o   = reinterpret_cast<v4f*>(out);
        v4f v = __builtin_nontemporal_load(&zin[gtid]);
        v4f r;
        r.x = i1e_eval(v.x);
        r.y = i1e_eval(v.y);
        r.z = i1e_eval(v.z);
        r.w = i1e_eval(v.w);
        __builtin_nontemporal_store(r, &o[gtid]);       // global_store_b128 NT
    }

    // Scalar tail (n % 4 elements) — empty for this shape (n divisible by 4),
    // kept for generality. First few lanes of block 0 handle it.
    const long long tail_base = n4 << 2;
    const long long tail_cnt  = n - tail_base;
    if (blockIdx.x == 0 && (long long)threadIdx.x < tail_cnt) {
        const long long i = tail_base + threadIdx.x;
        float v = __builtin_nontemporal_load(&z[i]);
        __builtin_nontemporal_store(i1e_eval(v), &out[i]);
    }
}

extern "C" void kernel_launch(void* const* d_in, const int* in_sizes, int n_in,
                              void* d_out, int out_size, void* d_ws, size_t ws_size,
                              hipStream_t stream) {
    (void)n_in; (void)d_ws; (void)ws_size; (void)out_size;
    const float* z   = reinterpret_cast<const float*>(d_in[0]);
    float*       out = reinterpret_cast<float*>(d_out);
    const long long n  = (long long)in_sizes[0];
    const long long n4 = n >> 2;

    const int block = 256;                              // 8 wave32 waves / block
    long long work = n4 > 0 ? n4 : 1;
    long long grid_ll = (work + block - 1) / block;
    if (grid_ll < 1) grid_ll = 1;
    const unsigned grid = (unsigned)grid_ll;

    ive_i1e_kernel<<<grid, block, 0, stream>>>(z, out, n);
}